// ObjectDetectionModelV3_40003325395675
// MI455X (gfx1250) — compile-verified
//
#include <hip/hip_runtime.h>
#include <hip/hip_bf16.h>
#include <math.h>

typedef float v2f __attribute__((ext_vector_type(2)));
typedef float v8f __attribute__((ext_vector_type(8)));

#define HW0 (512 * 512)

// ---------------------------------------------------------------------------
// Kernel 0: zero the 16x480 descriptor accumulator (stored as u32 bit patterns)
// ---------------------------------------------------------------------------
__global__ void k_zero(unsigned* __restrict__ p, int n) {
    int i = blockIdx.x * blockDim.x + threadIdx.x;
    if (i < n) p[i] = 0u;
}

// ---------------------------------------------------------------------------
// Kernel 1: per-pixel argmax over 16 channels + verbatim copy of `encoded`
// (second tuple output). mask0[p] = 1u << argmax_label. Coalesced per channel.
// ---------------------------------------------------------------------------
__global__ void k_argmax_copy(const float* __restrict__ enc,
                              float* __restrict__ enc_out,
                              unsigned* __restrict__ mask0) {
    int p = blockIdx.x * blockDim.x + threadIdx.x;
    if (p >= HW0) return;
    float best = enc[p];
    enc_out[p] = best;
    int bi = 0;
#pragma unroll
    for (int c = 1; c < 16; ++c) {
        float v = enc[c * HW0 + p];
        enc_out[c * HW0 + p] = v;
        if (v > best) { best = v; bi = c; }   // strict >: first-max tiebreak like jnp.argmax
    }
    mask0[p] = 1u << bi;
}

// ---------------------------------------------------------------------------
// Kernel 2: 2x2 OR-pool of label bitmasks (== 2x2 max-pool of binary masks)
// ---------------------------------------------------------------------------
__global__ void k_pool_or(const unsigned* __restrict__ in,
                          unsigned* __restrict__ out, int So) {
    int idx = blockIdx.x * blockDim.x + threadIdx.x;
    if (idx >= So * So) return;
    int i = idx / So, j = idx - i * So;
    int Si = So * 2;
    const unsigned* r0 = in + (2 * i) * Si + 2 * j;
    const unsigned* r1 = r0 + Si;
    out[idx] = r0[0] | r0[1] | r1[0] | r1[1];
}

// ---------------------------------------------------------------------------
// Kernel 3: per-(label, channel) masked spatial max.
// vec[k][colBase + c] = max over pixels of (mask bit k set ? feat : 0), >= 0.
// Wave32 butterfly reduce, then integer atomicMax on float bits (valid: >=0).
// ---------------------------------------------------------------------------
__global__ void k_masked_max(const float* __restrict__ feat,
                             const unsigned* __restrict__ mask,
                             unsigned* __restrict__ vecbits,
                             int S2, int colBase, int blocksPerChannel) {
    int c      = blockIdx.x / blocksPerChannel;
    int chunk  = blockIdx.x - c * blocksPerChannel;
    int pxPer  = S2 / blocksPerChannel;
    int p0     = chunk * pxPer;
    const float* f = feat + (size_t)c * (size_t)S2;

    float r[16];
#pragma unroll
    for (int k = 0; k < 16; ++k) r[k] = 0.0f;

    for (int p = p0 + (int)threadIdx.x; p < p0 + pxPer; p += (int)blockDim.x) {
        __builtin_prefetch(&f[p + 4 * (int)blockDim.x], 0, 0);  // global_prefetch_b8
        unsigned m = mask[p];   // tiny array; stays L2-resident across channels
        float    v = f[p];
#pragma unroll
        for (int k = 0; k < 16; ++k)
            if (m & (1u << k)) r[k] = fmaxf(r[k], v);
    }

    // wave32 butterfly max
#pragma unroll
    for (int k = 0; k < 16; ++k) {
#pragma unroll
        for (int off = 16; off > 0; off >>= 1)
            r[k] = fmaxf(r[k], __shfl_xor(r[k], off, 32));
    }

    __shared__ float red[8][16];
    int wave = threadIdx.x >> 5, lane = threadIdx.x & 31;
    if (lane == 0) {
#pragma unroll
        for (int k = 0; k < 16; ++k) red[wave][k] = r[k];
    }
    __syncthreads();
    int nw = blockDim.x >> 5;
    if ((int)threadIdx.x < 16) {
        float m = red[0][threadIdx.x];
        for (int w = 1; w < nw; ++w) m = fmaxf(m, red[w][threadIdx.x]);
        atomicMax(&vecbits[threadIdx.x * 480 + colBase + c], __float_as_uint(m));
    }
}

// ---------------------------------------------------------------------------
// Kernel 4: MLP head on the WMMA pipe. vec(16x480) @ W1(480x240) + b1 -> h,
// h @ W2(240x4, zero-padded to N=16) + b2 -> sigmoid -> out rows 1..15.
// f32 WMMA 16x16x4; A fragment: lanes 0-15 hold K={k0,k0+1}, lanes 16-31
// K={k0+2,k0+3}; B fragment mirrors with rows striped across lanes;
// C/D: vgpr v -> row v (lanes 0-15) / v+8 (lanes 16-31), col = lane&15.
// ---------------------------------------------------------------------------
__global__ __launch_bounds__(512, 1) void k_mlp(const float* __restrict__ vecf,
                                                const float* __restrict__ W1,
                                                const float* __restrict__ b1,
                                                const float* __restrict__ W2,
                                                const float* __restrict__ b2,
                                                float* __restrict__ out) {
    __shared__ float vs[16][480];
    __shared__ float hbuf[16][240];
    int tid = threadIdx.x;
    for (int i = tid; i < 16 * 480; i += 512) vs[i / 480][i % 480] = vecf[i];
    __syncthreads();

    int wave = tid >> 5;
    int lane = tid & 31;
    int hlf  = lane >> 4;   // 0: lanes 0-15, 1: lanes 16-31
    int l16  = lane & 15;

#if __has_builtin(__builtin_amdgcn_wmma_f32_16x16x4_f32)
    if (wave < 15) {                 // wave-uniform branch: EXEC stays all-ones
        int n = wave * 16 + l16;
        v8f acc = {};
        for (int k0 = 0; k0 < 480; k0 += 4) {
            int ka = k0 + 2 * hlf;
            v2f a, b;
            a.x = vs[l16][ka];
            a.y = vs[l16][ka + 1];
            b.x = W1[ka * 240 + n];
            b.y = W1[(ka + 1) * 240 + n];
            acc = __builtin_amdgcn_wmma_f32_16x16x4_f32(
                false, a, false, b, (short)0, acc, false, false);
        }
        float bb = b1[n];
#pragma unroll
        for (int v = 0; v < 8; ++v) hbuf[v + 8 * hlf][n] = acc[v] + bb;
    }
    __syncthreads();
    if (wave == 0) {
        v8f acc = {};
        for (int k0 = 0; k0 < 240; k0 += 4) {
            int ka = k0 + 2 * hlf;
            v2f a, b;
            a.x = hbuf[l16][ka];
            a.y = hbuf[l16][ka + 1];
            b.x = (l16 < 4) ? W2[ka * 4 + l16] : 0.0f;         // pad N to 16
            b.y = (l16 < 4) ? W2[(ka + 1) * 4 + l16] : 0.0f;
            acc = __builtin_amdgcn_wmma_f32_16x16x4_f32(
                false, a, false, b, (short)0, acc, false, false);
        }
        if (l16 < 4) {
            float bb = b2[l16];
#pragma unroll
            for (int v = 0; v < 8; ++v) {
                int m = v + 8 * hlf;
                if (m >= 1) {
                    float x = acc[v] + bb;
                    out[(m - 1) * 4 + l16] = 1.0f / (1.0f + expf(-x));
                }
            }
        }
    }
#else
    // Scalar fallback (should not be taken on gfx1250 toolchains)
    for (int i = tid; i < 16 * 240; i += 512) {
        int m = i / 240, n = i % 240;
        float s = b1[n];
        for (int k = 0; k < 480; ++k) s += vs[m][k] * W1[k * 240 + n];
        hbuf[m][n] = s;
    }
    __syncthreads();
    for (int i = tid; i < 16 * 4; i += 512) {
        int m = i / 4, n = i % 4;
        float s = b2[n];
        for (int k = 0; k < 240; ++k) s += hbuf[m][k] * W2[k * 4 + n];
        if (m >= 1) out[(m - 1) * 4 + n] = 1.0f / (1.0f + expf(-s));
    }
#endif
}

// ---------------------------------------------------------------------------
extern "C" void kernel_launch(void* const* d_in, const int* in_sizes, int n_in,
                              void* d_out, int out_size, void* d_ws, size_t ws_size,
                              hipStream_t stream) {
    const float* enc = (const float*)d_in[0];
    const float* f0  = (const float*)d_in[1];
    const float* f1  = (const float*)d_in[2];
    const float* f2  = (const float*)d_in[3];
    const float* f3  = (const float*)d_in[4];
    const float* W1  = (const float*)d_in[5];
    const float* b1  = (const float*)d_in[6];
    const float* W2  = (const float*)d_in[7];
    const float* b2  = (const float*)d_in[8];

    float* out_bbox = (float*)d_out;       // 60 floats: (1, 15, 4)
    float* out_enc  = out_bbox + 60;       // encoded passthrough (4,194,304 floats)

    unsigned* mask0   = (unsigned*)d_ws;        // 512*512  = 262144 u32
    unsigned* mask1   = mask0 + 262144;         // 256*256  =  65536 u32
    unsigned* mask2   = mask1 + 65536;          // 128*128  =  16384 u32
    unsigned* mask3   = mask2 + 16384;          //  64*64   =   4096 u32
    unsigned* vecbits = mask3 + 4096;           // 16*480   =   7680 u32 (float bits)

    k_zero<<<(7680 + 255) / 256, 256, 0, stream>>>(vecbits, 7680);
    k_argmax_copy<<<(HW0 + 255) / 256, 256, 0, stream>>>(enc, out_enc, mask0);
    k_pool_or<<<(256 * 256 + 255) / 256, 256, 0, stream>>>(mask0, mask1, 256);
    k_pool_or<<<(128 * 128 + 255) / 256, 256, 0, stream>>>(mask1, mask2, 128);
    k_pool_or<<<(64 * 64 + 255) / 256, 256, 0, stream>>>(mask2, mask3, 64);

    // (feat, mask, vec, S^2, colBase, blocksPerChannel)
    k_masked_max<<<32 * 32, 256, 0, stream>>>(f0, mask0, vecbits, 262144, 0, 32);
    k_masked_max<<<64 * 8,  256, 0, stream>>>(f1, mask1, vecbits, 65536, 32, 8);
    k_masked_max<<<128 * 4, 256, 0, stream>>>(f2, mask2, vecbits, 16384, 96, 4);
    k_masked_max<<<256 * 1, 256, 0, stream>>>(f3, mask3, vecbits, 4096, 224, 1);

    k_mlp<<<1, 512, 0, stream>>>((const float*)vecbits, W1, b1, W2, b2, out_bbox);
}